// PGATCrossAttentionLayer_15530601742961
// MI455X (gfx1250) — compile-verified
//
#include <hip/hip_runtime.h>
#include <math.h>

// PGAT cross-attention for MI455X (gfx1250, wave32).
// Algebraic restructuring: keys/values are never materialized; query is
// back-projected through Wak/Wok ("p" vectors), attention mixes RAW value
// features ("y"), and Wav@Wpa is pre-folded into Wcomb. 90 GFLOP -> 9.3 GFLOP,
// ~270 MB traffic -> memory-bound at 23.3 TB/s (~12us floor).
// All GEMM stages use V_WMMA_F32_16X16X4_F32 (full f32 precision).
// Workspace requirement: ~119 MB (written fully each call).

typedef __attribute__((ext_vector_type(2))) float v2f;
typedef __attribute__((ext_vector_type(4))) float v4f;
typedef __attribute__((ext_vector_type(8))) float v8f;

#define NPTS 4096
#define KNB  16
#define KOB  8
#define HH   8
#define CC   128
#define HCDIM 1024
#define SCALEF 0.25f   // (C // H)^-0.5 = 16^-0.5

// ---------------------------------------------------------------------------
// Generic strided/batched f32 WMMA GEMM:  C[m,n] = sum_k A(m,k)*B(k,n) + bias[n]
// A(m,k) = A[z*bsA + m*sAm + k*sAk],  B(k,n) = B[z*bsB + k*sBk + n*sBn]
// C at C[z*bsC + m*ldc + n].  Requires M%64==0, N%64==0, K%32==0.
// Workgroup tile 64x64: 4 waves stacked in M, each wave = 16(M) x 64(N)
// via four 16x16 f32 accumulators; K staged through LDS in chunks of 32.
// Both tiles stored K-contiguous with stride KC+2 = 34 floats:
//   - rows are 8B aligned (34*4=136), even k offsets are 8B aligned
//   - frag read = one aligned ds_load_b64 into an even VGPR pair (no shuffles)
//   - bank check (8B units, 32 bank-pairs): addr8 = 17*row + k/2; 17 is
//     invertible mod 32 and 17^-1 = 17, so the 16+16 lane pattern never
//     collides -> conflict-free.
// ---------------------------------------------------------------------------
#define KC 32
#define LDP (KC + 2)
__global__ __launch_bounds__(128)
void wmma_gemm_f32(const float* __restrict__ A, const float* __restrict__ B,
                   const float* __restrict__ bias, float* __restrict__ C,
                   int Kdim,
                   long sAm, long sAk, long bsA,
                   long sBk, long sBn, long bsB,
                   long ldc, long bsC)
{
    __shared__ __align__(16) float Alds[64][LDP];   // [m][k]
    __shared__ __align__(16) float Blds[64][LDP];   // [n][k]  (K-contiguous!)

    const int wave = threadIdx.x >> 5;      // 0..3 -> M sub-block
    const int lane = threadIdx.x & 31;
    const int m0 = blockIdx.y * 64;
    const int n0 = blockIdx.x * 64;
    const float* Ab = A + (long)blockIdx.z * bsA;
    const float* Bb = B + (long)blockIdx.z * bsB;
    float* Cb = C + (long)blockIdx.z * bsC;

    v8f acc[4] = {};

    const int row  = lane & 15;
    const int koff = (lane >> 4) << 1;   // lanes 0-15: K=0,1 ; lanes 16-31: K=2,3

    for (int kb = 0; kb < Kdim; kb += KC) {
        // A staging: all our A operands have sAk==1 -> k-major mapping is
        // coalesced in global and sequential in LDS.
        for (int i = threadIdx.x; i < 64 * KC; i += 128) {
            int k = i & (KC - 1);
            int m = i >> 5;
            Alds[m][k] = Ab[(long)(m0 + m) * sAm + (long)(kb + k) * sAk];
        }
        // B staging: pick mapping so global reads stay coalesced.
        if (sBn == 1) {          // row-major B: consecutive n contiguous
            for (int i = threadIdx.x; i < 64 * KC; i += 128) {
                int n = i & 63;
                int k = i >> 6;
                Blds[n][k] = Bb[(long)(kb + k) * sBk + (long)(n0 + n)];
            }
        } else {                 // transposed B (sBk==1): consecutive k contiguous
            for (int i = threadIdx.x; i < 64 * KC; i += 128) {
                int k = i & (KC - 1);
                int n = i >> 5;
                Blds[n][k] = Bb[(long)(kb + k) + (long)(n0 + n) * sBn];
            }
        }
        __syncthreads();

#pragma unroll
        for (int ks = 0; ks < KC; ks += 4) {
            const int k0 = ks + koff;
            v2f a = *(const v2f*)&Alds[wave * 16 + row][k0];
#pragma unroll
            for (int j = 0; j < 4; ++j) {
                v2f b = *(const v2f*)&Blds[j * 16 + row][k0];
                acc[j] = __builtin_amdgcn_wmma_f32_16x16x4_f32(
                    false, a, false, b, (short)0, acc[j], false, false);
            }
        }
        __syncthreads();
    }

    // 16x16 f32 D layout: VGPR r holds M=r (lanes 0-15) / M=8+r (lanes 16-31).
    const int mbase = m0 + wave * 16 + ((lane >> 4) << 3);
#pragma unroll
    for (int j = 0; j < 4; ++j) {
        const int n = n0 + j * 16 + row;
        const float bv = bias ? bias[n] : 0.0f;
#pragma unroll
        for (int r = 0; r < 8; ++r)
            Cb[(long)(mbase + r) * ldc + n] = acc[j][r] + bv;
    }
}

// ---------------------------------------------------------------------------
// Fused attention core, one block per query row n:
//  - distance decay weights (agents clamp dist at 0.001, obstacles don't)
//  - logits dot[h,k] = x_k . p[n,h] + q[n,h].b_key  (times SCALE * dw)
//  - softmax over k
//  - y[n,h,:] = sum_k att[h,k] * raw_value_features[k,:]
// Streams the 168 MB of raw features exactly once with b128 accesses.
// Shared rows padded to 260/132 floats: 16B aligned, stride == 4 (mod 64)
// so the k-strided 16B dot-product reads hit disjoint bank quads.
// ---------------------------------------------------------------------------
#define LAP 260   // 256 + 4
#define LOP 132   // 128 + 4
__global__ __launch_bounds__(256)
void attend_fused(const float* __restrict__ xak, const float* __restrict__ xav,
                  const float* __restrict__ xok, const float* __restrict__ xov,
                  const float* __restrict__ apos, const float* __restrict__ npos,
                  const float* __restrict__ opos,
                  const float* __restrict__ Q,
                  const float* __restrict__ Pa, const float* __restrict__ Po,
                  const float* __restrict__ bak, const float* __restrict__ bok,
                  const float* __restrict__ decayA, const float* __restrict__ decayO,
                  float* __restrict__ Ya, float* __restrict__ Yo)
{
    const int n = blockIdx.x;
    const int t = threadIdx.x;

    __shared__ __align__(16) float sak[KNB][LAP];
    __shared__ __align__(16) float sav[KNB][LAP];
    __shared__ __align__(16) float spa[HH][LAP];
    __shared__ __align__(16) float sok[KOB][LOP];
    __shared__ __align__(16) float sov[KOB][LOP];
    __shared__ __align__(16) float spo[HH][LOP];
    __shared__ float satta[HH][KNB];
    __shared__ float satto[HH][KOB];
    __shared__ float sdwa[KNB];
    __shared__ float sdwo[KOB];
    __shared__ float sqba[HH];
    __shared__ float sqbo[HH];

    const v4f* xakn = (const v4f*)(xak + (size_t)n * KNB * 256);
    const v4f* xavn = (const v4f*)(xav + (size_t)n * KNB * 256);
    const v4f* xokn = (const v4f*)(xok + (size_t)n * KOB * 128);
    const v4f* xovn = (const v4f*)(xov + (size_t)n * KOB * 128);
    const v4f* pan  = (const v4f*)(Pa + (size_t)n * 2048);
    const v4f* pon  = (const v4f*)(Po + (size_t)n * 1024);

    for (int i = t; i < KNB * 64; i += 256) {       // 1024 x b128
        int k = i >> 6, d4 = (i & 63) << 2;
        *(v4f*)&sak[k][d4] = xakn[i];
        *(v4f*)&sav[k][d4] = xavn[i];
    }
    for (int i = t; i < HH * 64; i += 256)          // 512 x b128
        *(v4f*)&spa[i >> 6][(i & 63) << 2] = pan[i];
    for (int i = t; i < KOB * 32; i += 256) {       // 256 x b128
        int k = i >> 5, d4 = (i & 31) << 2;
        *(v4f*)&sok[k][d4] = xokn[i];
        *(v4f*)&sov[k][d4] = xovn[i];
    }
    for (int i = t; i < HH * 32; i += 256)
        *(v4f*)&spo[i >> 5][(i & 31) << 2] = pon[i];

    const float ax = apos[n * 3 + 0], ay = apos[n * 3 + 1], az = apos[n * 3 + 2];
    if (t < KNB) {
        float dx = ax - npos[(n * KNB + t) * 3 + 0];
        float dy = ay - npos[(n * KNB + t) * 3 + 1];
        float dz = az - npos[(n * KNB + t) * 3 + 2];
        float dist = fmaxf(sqrtf(dx * dx + dy * dy + dz * dz), 0.001f);
        sdwa[t] = expf(-decayA[0] * dist);
    }
    if (t >= 32 && t < 32 + KOB) {
        int k = t - 32;
        float dx = ax - opos[(n * KOB + k) * 3 + 0];
        float dy = ay - opos[(n * KOB + k) * 3 + 1];
        float dz = az - opos[(n * KOB + k) * 3 + 2];
        float dist = sqrtf(dx * dx + dy * dy + dz * dz);   // no clamp
        sdwo[k] = expf(-decayO[0] * dist);
    }
    if (t >= 64 && t < 64 + HH) {   // q . key-bias constants
        int h = t - 64;
        float sa = 0.f, so = 0.f;
        for (int c = 0; c < CC; ++c) {
            float q = Q[(size_t)n * HCDIM + h * CC + c];
            sa += q * bak[h * CC + c];
            so += q * bok[h * CC + c];
        }
        sqba[h] = sa; sqbo[h] = so;
    }
    __syncthreads();

    if (t < HH * KNB) {                       // 128 agent logits, v4f dots
        int h = t >> 4, k = t & 15;
        const v4f* ak = (const v4f*)sak[k];
        const v4f* pa = (const v4f*)spa[h];
        v4f vs = {};
        for (int d = 0; d < 64; ++d) vs += ak[d] * pa[d];
        float s = sqba[h] + vs[0] + vs[1] + vs[2] + vs[3];
        satta[h][k] = s * SCALEF * sdwa[k];
    } else if (t < HH * KNB + HH * KOB) {     // 64 obstacle logits
        int u = t - HH * KNB;
        int h = u >> 3, k = u & 7;
        const v4f* ok = (const v4f*)sok[k];
        const v4f* po = (const v4f*)spo[h];
        v4f vs = {};
        for (int d = 0; d < 32; ++d) vs += ok[d] * po[d];
        float s = sqbo[h] + vs[0] + vs[1] + vs[2] + vs[3];
        satto[h][k] = s * SCALEF * sdwo[k];
    }
    __syncthreads();

    if (t < HH) {
        float m = -3.4e38f;
        for (int k = 0; k < KNB; ++k) m = fmaxf(m, satta[t][k]);
        float sum = 0.f;
        for (int k = 0; k < KNB; ++k) { float e = expf(satta[t][k] - m); satta[t][k] = e; sum += e; }
        float inv = 1.f / sum;
        for (int k = 0; k < KNB; ++k) satta[t][k] *= inv;
    } else if (t >= 32 && t < 32 + HH) {
        int h = t - 32;
        float m = -3.4e38f;
        for (int k = 0; k < KOB; ++k) m = fmaxf(m, satto[h][k]);
        float sum = 0.f;
        for (int k = 0; k < KOB; ++k) { float e = expf(satto[h][k] - m); satto[h][k] = e; sum += e; }
        float inv = 1.f / sum;
        for (int k = 0; k < KOB; ++k) satto[h][k] *= inv;
    }
    __syncthreads();

    v4f* Yan = (v4f*)(Ya + (size_t)n * 2048);
    v4f* Yon = (v4f*)(Yo + (size_t)n * 1024);
    for (int i = t; i < HH * 64; i += 256) {        // 512 x b128 out
        int h = i >> 6, d4 = (i & 63) << 2;
        v4f s = {};
        for (int k = 0; k < KNB; ++k) s += satta[h][k] * *(const v4f*)&sav[k][d4];
        Yan[i] = s;
    }
    for (int i = t; i < HH * 32; i += 256) {        // 256 x b128 out
        int h = i >> 5, d4 = (i & 31) << 2;
        v4f s = {};
        for (int k = 0; k < KOB; ++k) s += satto[h][k] * *(const v4f*)&sov[k][d4];
        Yon[i] = s;
    }
}

// bcomb = b_value @ Wproj + b_proj (softmax weights sum to 1 -> bias folds through)
__global__ __launch_bounds__(128)
void bcomb_kernel(const float* __restrict__ bav, const float* __restrict__ Wpa,
                  const float* __restrict__ bpa,
                  const float* __restrict__ bov, const float* __restrict__ Wpo,
                  const float* __restrict__ bpo,
                  float* __restrict__ bca, float* __restrict__ bco)
{
    int j = threadIdx.x;  // 128 output channels
    float sa = bpa[j], so = bpo[j];
    for (int i = 0; i < HCDIM; ++i) {
        sa += bav[i] * Wpa[i * CC + j];
        so += bov[i] * Wpo[i * CC + j];
    }
    bca[j] = sa; bco[j] = so;
}

extern "C" void kernel_launch(void* const* d_in, const int* in_sizes, int n_in,
                              void* d_out, int out_size, void* d_ws, size_t ws_size,
                              hipStream_t stream)
{
    (void)in_sizes; (void)n_in; (void)out_size; (void)ws_size;
    const float* qf   = (const float*)d_in[0];
    const float* xak  = (const float*)d_in[1];
    const float* xav  = (const float*)d_in[2];
    const float* xok  = (const float*)d_in[3];
    const float* xov  = (const float*)d_in[4];
    const float* apos = (const float*)d_in[5];
    const float* npos = (const float*)d_in[6];
    const float* opos = (const float*)d_in[7];
    const float* Wq   = (const float*)d_in[8];
    const float* bq   = (const float*)d_in[9];
    const float* Wak  = (const float*)d_in[10];
    const float* bak  = (const float*)d_in[11];
    const float* Wav  = (const float*)d_in[12];
    const float* bav  = (const float*)d_in[13];
    const float* Wok  = (const float*)d_in[14];
    const float* bok  = (const float*)d_in[15];
    const float* Wov  = (const float*)d_in[16];
    const float* bov  = (const float*)d_in[17];
    const float* Wpa  = (const float*)d_in[18];
    const float* bpa  = (const float*)d_in[19];
    const float* Wpo  = (const float*)d_in[20];
    const float* bpo  = (const float*)d_in[21];
    const float* dA   = (const float*)d_in[22];
    const float* dO   = (const float*)d_in[23];
    float* out = (float*)d_out;

    float* ws    = (float*)d_ws;
    float* wsQ   = ws;                                 // [4096,1024]
    float* wsPa  = wsQ  + (size_t)NPTS * HCDIM;        // [4096,8,256]
    float* wsPo  = wsPa + (size_t)NPTS * 2048;         // [4096,8,128]
    float* wsYa  = wsPo + (size_t)NPTS * 1024;         // [4096,8,256]
    float* wsYo  = wsYa + (size_t)NPTS * 2048;         // [4096,8,128]
    float* wsWca = wsYo + (size_t)NPTS * 1024;         // [2048,128]
    float* wsWco = wsWca + (size_t)2048 * 128;         // [1024,128]
    float* wsBca = wsWco + (size_t)1024 * 128;         // [128]
    float* wsBco = wsBca + 128;                        // [128]

    // 1) Q = qf @ Wq + bq   [4096,256]x[256,1024]
    wmma_gemm_f32<<<dim3(HCDIM / 64, NPTS / 64, 1), 128, 0, stream>>>(
        qf, Wq, bq, wsQ, 256,  256, 1, 0,  1024, 1, 0,  1024, 0);

    // 2) Pa_h = Q_h @ Wak_h^T   batched over h: [4096,128]x[128,256]
    wmma_gemm_f32<<<dim3(256 / 64, NPTS / 64, HH), 128, 0, stream>>>(
        wsQ, Wak, nullptr, wsPa, 128,  1024, 1, 128,  1, 1024, 128,  2048, 256);

    // 3) Po_h = Q_h @ Wok_h^T   [4096,128]x[128,128]
    wmma_gemm_f32<<<dim3(128 / 64, NPTS / 64, HH), 128, 0, stream>>>(
        wsQ, Wok, nullptr, wsPo, 128,  1024, 1, 128,  1, 1024, 128,  1024, 128);

    // 4) Wcomb_a[h] = Wav_h @ Wpa_h   [256,128]x[128,128] x8
    wmma_gemm_f32<<<dim3(128 / 64, 256 / 64, HH), 128, 0, stream>>>(
        Wav, Wpa, nullptr, wsWca, 128,  1024, 1, 128,  128, 1, 128 * 128,  128, 256 * 128);

    // 5) Wcomb_o[h] = Wov_h @ Wpo_h   [128,128]x[128,128] x8
    wmma_gemm_f32<<<dim3(128 / 64, 128 / 64, HH), 128, 0, stream>>>(
        Wov, Wpo, nullptr, wsWco, 128,  1024, 1, 128,  128, 1, 128 * 128,  128, 128 * 128);

    // 6) folded biases
    bcomb_kernel<<<1, 128, 0, stream>>>(bav, Wpa, bpa, bov, Wpo, bpo, wsBca, wsBco);

    // 7) fused distance-decay attention -> mixed raw-feature vectors Y
    attend_fused<<<NPTS, 256, 0, stream>>>(
        xak, xav, xok, xov, apos, npos, opos,
        wsQ, wsPa, wsPo, bak, bok, dA, dO, wsYa, wsYo);

    // 8) agent_attended = Ya @ Wcomb_a + bcomb_a   [4096,2048]x[2048,128]
    wmma_gemm_f32<<<dim3(128 / 64, NPTS / 64, 1), 128, 0, stream>>>(
        wsYa, wsWca, wsBca, out, 2048,  2048, 1, 0,  128, 1, 0,  128, 0);

    // 9) obstacle_attended = Yo @ Wcomb_o + bcomb_o  [4096,1024]x[1024,128]
    wmma_gemm_f32<<<dim3(128 / 64, NPTS / 64, 1), 128, 0, stream>>>(
        wsYo, wsWco, wsBco, out + (size_t)NPTS * CC, 1024,  1024, 1, 0,  128, 1, 0,  128, 0);
}